// L2Similarity_20873541059196
// MI455X (gfx1250) — compile-verified
//
#include <hip/hip_runtime.h>

typedef float v2f __attribute__((ext_vector_type(2)));
typedef float v4f __attribute__((ext_vector_type(4)));
typedef float v8f __attribute__((ext_vector_type(8)));

#define D_COLS 256

// ---------------------------------------------------------------------------
// Stage 1: streaming partial reduction. grid = nb blocks x 256 threads.
// Thread t handles float4-column (t&63) of rows {blk*4 + (t>>6) + k*nb*4}.
// Produces per-block: colsumA[256], colsumB[256], sqA, sqB.
// Loads are 16B-coalesced (512B contiguous per wave) and non-temporal
// (stream-once data; don't thrash L2 even though it is 192MB).
// ---------------------------------------------------------------------------
__global__ __launch_bounds__(256) void l2sim_partial(
    const float* __restrict__ A, const float* __restrict__ B,
    float* __restrict__ colA, float* __restrict__ colB,
    float* __restrict__ sqA, float* __restrict__ sqB,
    int n1, int n2, int nb)
{
    const int t  = threadIdx.x;
    const int g  = t >> 6;   // row group within block: 0..3
    const int c4 = t & 63;   // float4 column index

    const v4f* __restrict__ A4 = (const v4f*)A;
    const v4f* __restrict__ B4 = (const v4f*)B;

    v4f accA = {0.f, 0.f, 0.f, 0.f};
    v4f accB = {0.f, 0.f, 0.f, 0.f};
    float sa = 0.f, sb = 0.f;

    const int r0   = blockIdx.x * 4 + g;
    const int step = nb * 4;
    for (int r = r0; r < n1; r += step) {
        v4f v = __builtin_nontemporal_load(&A4[r * (D_COLS / 4) + c4]);
        accA += v;
        sa += v.x * v.x + v.y * v.y + v.z * v.z + v.w * v.w;
    }
    for (int r = r0; r < n2; r += step) {
        v4f v = __builtin_nontemporal_load(&B4[r * (D_COLS / 4) + c4]);
        accB += v;
        sb += v.x * v.x + v.y * v.y + v.z * v.z + v.w * v.w;
    }

    __shared__ v4f lds4[256];
    __shared__ v2f ldss[256];

    v2f p; p.x = sa; p.y = sb;
    ldss[t] = p;
    lds4[t] = accA;
    __syncthreads();
    if (t < 64) {
        v4f s = lds4[t] + lds4[t + 64] + lds4[t + 128] + lds4[t + 192];
        *(v4f*)(colA + (size_t)blockIdx.x * D_COLS + t * 4) = s;
    }
    __syncthreads();
    lds4[t] = accB;
    __syncthreads();
    if (t < 64) {
        v4f s = lds4[t] + lds4[t + 64] + lds4[t + 128] + lds4[t + 192];
        *(v4f*)(colB + (size_t)blockIdx.x * D_COLS + t * 4) = s;
    }
    for (int s = 128; s > 0; s >>= 1) {
        __syncthreads();
        if (t < s) ldss[t] += ldss[t + s];
    }
    if (t == 0) {
        sqA[blockIdx.x] = ldss[0].x;
        sqB[blockIdx.x] = ldss[0].y;
    }
}

// ---------------------------------------------------------------------------
// Stage 2 (1 block, 256 threads): reduce block partials deterministically,
// then compute cross = dot(colsumA, colsumB) with V_WMMA_F32_16X16X4_F32
// on wave 0:  P = U * V^T, U[m][k]=sA[16m+k], V[n][k]=sB[16n+k];
// cross = trace(P), accumulated over 4 chained K=4 WMMAs.
// Fragment layouts per CDNA5 ISA 7.12.2:
//   A 16x4 f32:  VGPR j, lane L -> M = L&15, K = j + 2*(L>>4)
//   B 4x16 f32:  VGPR j, lane L -> K = j + 2*(L>>4), N = L&15
//   C/D 16x16:   VGPR r, lane L -> M = r + 8*(L>>4), N = L&15
//   diag: lanes 0-7 -> acc[lane]; lanes 24-31 -> acc[lane-24]
// ---------------------------------------------------------------------------
__global__ __launch_bounds__(256) void l2sim_final(
    const float* __restrict__ colA, const float* __restrict__ colB,
    const float* __restrict__ sqA, const float* __restrict__ sqB,
    float* __restrict__ out,
    int nb, float n1f, float n2f, float inv_d)
{
    const int t  = threadIdx.x;
    const int g  = t >> 6;
    const int c4 = t & 63;

    const v4f* __restrict__ cA4 = (const v4f*)colA;
    const v4f* __restrict__ cB4 = (const v4f*)colB;

    v4f accA = {0.f, 0.f, 0.f, 0.f};
    v4f accB = {0.f, 0.f, 0.f, 0.f};
    for (int b = g; b < nb; b += 4) {
        accA += cA4[b * (D_COLS / 4) + c4];
        accB += cB4[b * (D_COLS / 4) + c4];
    }

    __shared__ v4f   lds4[256];
    __shared__ float sA[D_COLS];
    __shared__ float sB[D_COLS];
    __shared__ v2f   ldss[256];
    __shared__ float diag[32];

    v2f sq = {0.f, 0.f};
    for (int i = t; i < nb; i += 256) { sq.x += sqA[i]; sq.y += sqB[i]; }
    ldss[t] = sq;

    lds4[t] = accA;
    __syncthreads();
    if (t < 64) {
        v4f s = lds4[t] + lds4[t + 64] + lds4[t + 128] + lds4[t + 192];
        *(v4f*)(sA + t * 4) = s;
    }
    __syncthreads();
    lds4[t] = accB;
    __syncthreads();
    if (t < 64) {
        v4f s = lds4[t] + lds4[t + 64] + lds4[t + 128] + lds4[t + 192];
        *(v4f*)(sB + t * 4) = s;
    }
    for (int s = 128; s > 0; s >>= 1) {
        __syncthreads();
        if (t < s) ldss[t] += ldss[t + s];
    }
    __syncthreads();

    if (t < 32) {  // wave 0 fully active -> EXEC all ones, WMMA-legal
        const int m    = t & 15;
        const int half = t >> 4;
        v8f acc = {0.f, 0.f, 0.f, 0.f, 0.f, 0.f, 0.f, 0.f};
#pragma unroll
        for (int k0 = 0; k0 < 16; k0 += 4) {
            const int base = 16 * m + k0 + 2 * half;
            v2f a, b;
            a.x = sA[base]; a.y = sA[base + 1];
            b.x = sB[base]; b.y = sB[base + 1];
            acc = __builtin_amdgcn_wmma_f32_16x16x4_f32(
                false, a, false, b, (short)0, acc, false, false);
        }
        const int want = (t < 8) ? t : ((t >= 24) ? (t - 24) : -1);
        float contrib = 0.f;
#pragma unroll
        for (int i = 0; i < 8; ++i)
            if (want == i) contrib = acc[i];
        diag[t] = contrib;
    }
    __syncthreads();

    if (t == 0) {
        float cross = 0.f;
#pragma unroll
        for (int i = 0; i < 32; ++i) cross += diag[i];
        out[0] = (n2f * ldss[0].x + n1f * ldss[0].y - 2.f * cross) * inv_d;
    }
}

extern "C" void kernel_launch(void* const* d_in, const int* in_sizes, int n_in,
                              void* d_out, int out_size, void* d_ws, size_t ws_size,
                              hipStream_t stream)
{
    (void)n_in; (void)out_size;

    const float* A = (const float*)d_in[0];
    const float* B = (const float*)d_in[1];
    float* out = (float*)d_out;

    const int D  = D_COLS;
    const int n1 = in_sizes[0] / D;
    const int n2 = in_sizes[1] / D;

    int nb = 256;  // stage-1 block count (balances HBM saturation vs stage-2 size)
    const size_t per_block = (size_t)(2 * D + 2) * sizeof(float);
    if ((size_t)nb * per_block > ws_size) {
        nb = (int)(ws_size / per_block);
        if (nb < 1) nb = 1;
    }

    float* colA = (float*)d_ws;                 // nb * 256
    float* colB = colA + (size_t)nb * D;        // nb * 256
    float* sqA  = colB + (size_t)nb * D;        // nb
    float* sqB  = sqA + nb;                     // nb

    l2sim_partial<<<dim3(nb), dim3(256), 0, stream>>>(
        A, B, colA, colB, sqA, sqB, n1, n2, nb);
    l2sim_final<<<dim3(1), dim3(256), 0, stream>>>(
        colA, colB, sqA, sqB, out, nb, (float)n1, (float)n2, 1.0f / (float)D);
}